// GraphAttentionConv2d_42941083025881
// MI455X (gfx1250) — compile-verified
//
#include <hip/hip_runtime.h>
#include <hip/hip_bf16.h>

// ---------------------------------------------------------------------------
// GATv2 + LayerNorm pipeline for MI455X (gfx1250, wave32).
//   B=2, C=128, N=50000 -> M=100000 nodes, H=2 heads x FH=64 = 128 feats, E=800000.
// GEMMs use V_WMMA_F32_16X16X4_F32 (fp32 matrix pipe), edge phases are
// L2-resident gather/scatter with f32 atomics, softmax-max via ordered-uint
// atomicMax.
// ---------------------------------------------------------------------------

#define C_DIM   128
#define OUT_DIM 128          // H*FH per weight matrix
#define N_NODES 50000
#define M_NODES 100000
#define E_EDGES 800000
#define NEG_GAT 0.2f
#define NEG_ACT 0.01f
#define LN_EPS  1e-5f
#define SM_EPS  1e-16f

typedef float v2f __attribute__((ext_vector_type(2)));
typedef float v8f __attribute__((ext_vector_type(8)));

// order-preserving float <-> uint mapping for atomicMax-based segment max
__device__ __forceinline__ unsigned ord_enc(float f) {
    unsigned u = __float_as_uint(f);
    return (u & 0x80000000u) ? ~u : (u | 0x80000000u);
}
__device__ __forceinline__ float ord_dec(unsigned u) {
    return __uint_as_float((u & 0x80000000u) ? (u ^ 0x80000000u) : ~u);
}
#define ORD_NEG_INF 0x007FFFFFu   // ord_enc(-inf)

// ---------------------------------------------------------------------------
// Kernel 0: initialize accumulator (d_out), denom, segmax.
// ---------------------------------------------------------------------------
__global__ __launch_bounds__(256)
void init_kernel(float* __restrict__ out, float* __restrict__ denom,
                 unsigned* __restrict__ segmax) {
    int t = blockIdx.x * 256 + threadIdx.x;
    if (t < M_NODES * OUT_DIM) out[t] = 0.0f;
    if (t < M_NODES * 2) {
        denom[t] = 0.0f;
        segmax[t] = ORD_NEG_INF;
    }
}

// ---------------------------------------------------------------------------
// Kernel 1: dual GEMM  h_l = xn@W_l+b_l,  h_r = xn@W_r+b_r
//   xn[m][c] = x[b*C*N + c*N + n],  m = b*N+n  (N divisible by 16/128 tiles,
//   but B-boundary handled per-row).
//   Workgroup: 512 threads = 16 waves.  Tile: 128 rows x 256 output cols
//   (waves 0..7 -> W_l col strips, waves 8..15 -> W_r col strips).
//   A tile (128 rows x 128 ch, f32) staged in 64KB LDS, layout [c][r].
//   WMMA: V_WMMA_F32_16X16X4_F32, K-loop of 32 steps, 8 M-tiles per wave.
//   A frag layout (ISA 7.12.2): lanes 0-15 hold K=k,k+1; lanes 16-31 K=k+2,k+3.
// ---------------------------------------------------------------------------
__global__ __launch_bounds__(512)
void gemm_dual_kernel(const float* __restrict__ x,
                      const float* __restrict__ Wl, const float* __restrict__ bl,
                      const float* __restrict__ Wr, const float* __restrict__ br,
                      float* __restrict__ hl, float* __restrict__ hr) {
    __shared__ float ldsA[C_DIM * 128];   // 64 KB: [c][r]

    const int tid  = threadIdx.x;
    const int lane = tid & 31;
    const int wave = tid >> 5;
    const int m0   = blockIdx.x * 128;

    // cooperative stage of the 128x128 activation tile (coalesced on rows)
    for (int idx = tid; idx < C_DIM * 128; idx += 512) {
        const int r = idx & 127;
        const int c = idx >> 7;
        const int m = m0 + r;
        float v = 0.0f;
        if (m < M_NODES) {
            const int b = m / N_NODES;
            const int n = m - b * N_NODES;
            v = x[((size_t)b * C_DIM + c) * N_NODES + n];
        }
        ldsA[c * 128 + r] = v;
    }
    __syncthreads();

    const bool is_r      = (wave >= 8);
    const float* __restrict__ W  = is_r ? Wr : Wl;
    const float* __restrict__ bb = is_r ? br : bl;
    float* __restrict__ Ho       = is_r ? hr : hl;
    const int j0   = (wave & 7) * 16;
    const int ml   = lane & 15;          // M / N position within fragment
    const int jj   = j0 + ml;            // output column this lane touches
    const int koff = (lane < 16) ? 0 : 2;

    v8f acc[8] = {};                     // 8 M-tiles x 16x16 f32 accumulators

    for (int k = 0; k < C_DIM; k += 4) {
        v2f bf;                          // B frag: B[k..k+3][j0..j0+15]
        bf.x = W[(k + koff)     * OUT_DIM + jj];
        bf.y = W[(k + koff + 1) * OUT_DIM + jj];
#pragma unroll
        for (int mt = 0; mt < 8; ++mt) {
            v2f af;                      // A frag from LDS
            af.x = ldsA[(k + koff)     * 128 + mt * 16 + ml];
            af.y = ldsA[(k + koff + 1) * 128 + mt * 16 + ml];
            acc[mt] = __builtin_amdgcn_wmma_f32_16x16x4_f32(
                false, af, false, bf, (short)0, acc[mt], false, false);
        }
    }

    // epilogue: +bias, store.  C/D layout: vgpr i -> M = i (+8 for upper half).
    const float bj = bb[jj];
#pragma unroll
    for (int mt = 0; mt < 8; ++mt) {
#pragma unroll
        for (int i = 0; i < 8; ++i) {
            const int row = m0 + mt * 16 + i + ((lane < 16) ? 0 : 8);
            if (row < M_NODES)
                Ho[(size_t)row * OUT_DIM + jj] = acc[mt][i] + bj;
        }
    }
}

// ---------------------------------------------------------------------------
// Kernel 2: per-edge GATv2 logits + segment max.  One wave per edge.
//   lane l handles feats [4l,4l+4); lanes 0-15 = head 0, 16-31 = head 1.
// ---------------------------------------------------------------------------
__global__ __launch_bounds__(256)
void edge_logits_kernel(const int* __restrict__ ei,
                        const float* __restrict__ hl, const float* __restrict__ hr,
                        const float* __restrict__ att,
                        float* __restrict__ logits, unsigned* __restrict__ segmax) {
    const int e    = (blockIdx.x * 256 + threadIdx.x) >> 5;
    const int lane = threadIdx.x & 31;
    if (e >= E_EDGES) return;

    const int src = ei[e];
    const int dst = ei[E_EDGES + e];

    const float4 a = *(const float4*)(hl + (size_t)src * OUT_DIM + lane * 4);
    const float4 b = *(const float4*)(hr + (size_t)dst * OUT_DIM + lane * 4);
    const float4 w = *(const float4*)(att + lane * 4);

    float t, s = 0.0f;
    t = a.x + b.x; s += w.x * (t > 0.0f ? t : NEG_GAT * t);
    t = a.y + b.y; s += w.y * (t > 0.0f ? t : NEG_GAT * t);
    t = a.z + b.z; s += w.z * (t > 0.0f ? t : NEG_GAT * t);
    t = a.w + b.w; s += w.w * (t > 0.0f ? t : NEG_GAT * t);

    // reduce within each 16-lane half (one head per half)
    s += __shfl_xor(s, 1, 32);
    s += __shfl_xor(s, 2, 32);
    s += __shfl_xor(s, 4, 32);
    s += __shfl_xor(s, 8, 32);

    if ((lane & 15) == 0) {
        const int h = lane >> 4;
        logits[(size_t)e * 2 + h] = s;
        atomicMax(&segmax[(size_t)dst * 2 + h], ord_enc(s));
    }
}

// ---------------------------------------------------------------------------
// Kernel 3: exp(logit - segmax[dst]) + denominator accumulation.
//   One thread per (edge, head); exp values overwrite the logits buffer.
// ---------------------------------------------------------------------------
__global__ __launch_bounds__(256)
void edge_exp_kernel(const int* __restrict__ ei, float* __restrict__ logits,
                     const unsigned* __restrict__ segmax, float* __restrict__ denom) {
    const int t = blockIdx.x * 256 + threadIdx.x;
    if (t >= E_EDGES * 2) return;
    const int e = t >> 1;
    const int h = t & 1;
    const int dst = ei[E_EDGES + e];
    const float m = ord_dec(segmax[(size_t)dst * 2 + h]);
    const float ex = expf(logits[t] - m);
    logits[t] = ex;
    atomicAdd(&denom[(size_t)dst * 2 + h], ex);
}

// ---------------------------------------------------------------------------
// Kernel 4: alpha-weighted scatter of h_l[src] into out[dst].  Wave per edge.
// ---------------------------------------------------------------------------
__global__ __launch_bounds__(256)
void edge_aggr_kernel(const int* __restrict__ ei, const float* __restrict__ hl,
                      const float* __restrict__ expl, const float* __restrict__ denom,
                      float* __restrict__ out) {
    const int e    = (blockIdx.x * 256 + threadIdx.x) >> 5;
    const int lane = threadIdx.x & 31;
    if (e >= E_EDGES) return;

    const int src = ei[e];
    const int dst = ei[E_EDGES + e];
    const int h   = lane >> 4;

    const float alpha = expl[(size_t)e * 2 + h] /
                        (denom[(size_t)dst * 2 + h] + SM_EPS);
    const float4 a = *(const float4*)(hl + (size_t)src * OUT_DIM + lane * 4);
    float* o = out + (size_t)dst * OUT_DIM + lane * 4;
    atomicAdd(o + 0, a.x * alpha);
    atomicAdd(o + 1, a.y * alpha);
    atomicAdd(o + 2, a.z * alpha);
    atomicAdd(o + 3, a.w * alpha);
}

// ---------------------------------------------------------------------------
// Kernel 5: +bias_gat, LayerNorm(128), LeakyReLU(0.01).  Wave per node,
//   full-wave shuffle reduction for mean / E[x^2]; in-place on d_out.
// ---------------------------------------------------------------------------
__global__ __launch_bounds__(256)
void ln_kernel(float* __restrict__ out, const float* __restrict__ bias,
               const float* __restrict__ gamma, const float* __restrict__ beta) {
    const int m    = (blockIdx.x * 256 + threadIdx.x) >> 5;
    const int lane = threadIdx.x & 31;
    if (m >= M_NODES) return;

    float4 v = *(float4*)(out + (size_t)m * OUT_DIM + lane * 4);
    const float4 bg = *(const float4*)(bias + lane * 4);
    v.x += bg.x; v.y += bg.y; v.z += bg.z; v.w += bg.w;

    float s  = v.x + v.y + v.z + v.w;
    float sq = v.x * v.x + v.y * v.y + v.z * v.z + v.w * v.w;
#pragma unroll
    for (int mask = 1; mask < 32; mask <<= 1) {
        s  += __shfl_xor(s,  mask, 32);
        sq += __shfl_xor(sq, mask, 32);
    }
    const float mean = s * (1.0f / OUT_DIM);
    const float var  = sq * (1.0f / OUT_DIM) - mean * mean;
    const float inv  = rsqrtf(var + LN_EPS);

    const float4 g  = *(const float4*)(gamma + lane * 4);
    const float4 bt = *(const float4*)(beta + lane * 4);
    float4 y;
    y.x = (v.x - mean) * inv * g.x + bt.x;
    y.y = (v.y - mean) * inv * g.y + bt.y;
    y.z = (v.z - mean) * inv * g.z + bt.z;
    y.w = (v.w - mean) * inv * g.w + bt.w;
    y.x = y.x > 0.0f ? y.x : NEG_ACT * y.x;
    y.y = y.y > 0.0f ? y.y : NEG_ACT * y.y;
    y.z = y.z > 0.0f ? y.z : NEG_ACT * y.z;
    y.w = y.w > 0.0f ? y.w : NEG_ACT * y.w;
    *(float4*)(out + (size_t)m * OUT_DIM + lane * 4) = y;
}

// ---------------------------------------------------------------------------
extern "C" void kernel_launch(void* const* d_in, const int* in_sizes, int n_in,
                              void* d_out, int out_size, void* d_ws, size_t ws_size,
                              hipStream_t stream) {
    const float* x     = (const float*)d_in[0];
    const int*   ei    = (const int*)d_in[1];
    const float* Wl    = (const float*)d_in[2];
    const float* bl    = (const float*)d_in[3];
    const float* Wr    = (const float*)d_in[4];
    const float* br    = (const float*)d_in[5];
    const float* att   = (const float*)d_in[6];
    const float* bias  = (const float*)d_in[7];
    const float* gamma = (const float*)d_in[8];
    const float* beta  = (const float*)d_in[9];
    float* out = (float*)d_out;

    // workspace carve-up (~110 MB)
    float* hl       = (float*)d_ws;                       // M*128
    float* hr       = hl + (size_t)M_NODES * OUT_DIM;     // M*128
    float* logits   = hr + (size_t)M_NODES * OUT_DIM;     // E*2 (reused for exp)
    float* denom    = logits + (size_t)E_EDGES * 2;       // M*2
    unsigned* segmax = (unsigned*)(denom + (size_t)M_NODES * 2); // M*2

    init_kernel<<<(M_NODES * OUT_DIM + 255) / 256, 256, 0, stream>>>(out, denom, segmax);
    gemm_dual_kernel<<<(M_NODES + 127) / 128, 512, 0, stream>>>(x, Wl, bl, Wr, br, hl, hr);
    edge_logits_kernel<<<(E_EDGES + 7) / 8, 256, 0, stream>>>(ei, hl, hr, att, logits, segmax);
    edge_exp_kernel<<<(E_EDGES * 2 + 255) / 256, 256, 0, stream>>>(ei, logits, segmax, denom);
    edge_aggr_kernel<<<(E_EDGES + 7) / 8, 256, 0, stream>>>(ei, hl, logits, denom, out);
    ln_kernel<<<(M_NODES + 7) / 8, 256, 0, stream>>>(out, bias, gamma, beta);
}